// patch_gaussian_28767690948688
// MI455X (gfx1250) — compile-verified
//
#include <hip/hip_runtime.h>
#include <hip/hip_bf16.h>

// Geometry (fixed by the reference)
#define BATCH   8
#define PATCHES 1024      // 32x32 patch grid
#define PN      32        // patches per side
#define G       32        // gaussians per patch
#define PS      16        // patch pixel size (PH == PW == 16)
#define IMG     512       // PN * PS
#define LOG2E   1.44269504088896340736f

typedef float v2f __attribute__((ext_vector_type(2)));
typedef float v8f __attribute__((ext_vector_type(8)));

__device__ __forceinline__ float sigmoid_precise(float x) {
    return 1.0f / (1.0f + expf(-x));
}

__global__ __launch_bounds__(256) void splat_wmma_kernel(
    const float* __restrict__ xyz,      // [B,P,G,2]
    const float* __restrict__ chol,     // [B,P,G,3]
    const float* __restrict__ color,    // [B,P,G,3]
    const float* __restrict__ opac,     // [B,P,G,1]
    float* __restrict__ out)            // [B,3,512,512]
{
    // Per-gaussian params: {mx, my, A=0.5*log2e*ca, B=log2e*cb, C=0.5*log2e*cc, op, 0, 0}
    __shared__ __align__(16) float sP[G * 8];
    // Per-gaussian color padded: {r, g, b, 0}
    __shared__ __align__(16) float sC[G * 4];

    const int patch = blockIdx.x;          // = b*1024 + p
    const int tid   = threadIdx.x;

    // ---------------- Phase 1: one wave computes gaussian params ----------------
    if (tid < G) {
        const int g = tid;
        const long base = (long)patch * G + g;

        const float x0 = xyz[base * 2 + 0];
        const float x1 = xyz[base * 2 + 1];
        const float c0 = chol[base * 3 + 0];
        const float c1 = chol[base * 3 + 1];
        const float c2 = chol[base * 3 + 2];
        const float cr = color[base * 3 + 0];
        const float cg = color[base * 3 + 1];
        const float cb_ = color[base * 3 + 2];
        const float op = opac[base];

        const float mx = (float)PS * sigmoid_precise(x0);
        const float my = (float)PS * sigmoid_precise(x1);

        const float l0 = fabsf(c0) + 0.5f;
        const float l1 = c1;
        const float l2 = fabsf(c2) + 0.5f;
        const float s00 = l0 * l0;
        const float s01 = l0 * l1;
        const float s11 = l1 * l1 + l2 * l2;
        const float det = s00 * s11 - s01 * s01;
        const float inv = 1.0f / det;           // det >= 0.0625, safe
        const float ca = s11 * inv;
        const float cbv = -s01 * inv;
        const float cc = s00 * inv;

        sP[g * 8 + 0] = mx;
        sP[g * 8 + 1] = my;
        sP[g * 8 + 2] = 0.5f * LOG2E * ca;
        sP[g * 8 + 3] = LOG2E * cbv;
        sP[g * 8 + 4] = 0.5f * LOG2E * cc;
        sP[g * 8 + 5] = sigmoid_precise(op);
        sP[g * 8 + 6] = 0.0f;
        sP[g * 8 + 7] = 0.0f;

        sC[g * 4 + 0] = cr;
        sC[g * 4 + 1] = cg;
        sC[g * 4 + 2] = cb_;
        sC[g * 4 + 3] = 0.0f;                   // pad channel -> zero cols N>=3
    }
    __syncthreads();

    // ---------------- Phase 2: WMMA accumulation over gaussians ----------------
    const int wave = tid >> 5;                  // 0..7, owns rows 2w and 2w+1
    const int lane = tid & 31;
    const int hf   = lane >> 4;                 // lane half (K-split)
    const int n    = lane & 15;                 // N column = color channel
    const int nc   = (n < 3) ? n : 3;           // clamp -> reads the zero pad

    const float px  = (float)n + 0.5f;          // M (pixel-x) == lane&15 for A
    const int   y0  = wave * 2;
    const int   y1  = y0 + 1;
    const float py0 = (float)y0 + 0.5f;
    const float py1 = (float)y1 + 0.5f;

    v8f acc0 = {0.f, 0.f, 0.f, 0.f, 0.f, 0.f, 0.f, 0.f};
    v8f acc1 = {0.f, 0.f, 0.f, 0.f, 0.f, 0.f, 0.f, 0.f};

    const float4* P4 = (const float4*)sP;

    // alpha(g, pixel) with log2e pre-folded: alpha = op * exp2(min(p2, 0))
    auto alphaf = [&](float4 Pa, float4 Pb, float py) -> float {
        const float dx = px - Pa.x;
        const float dy = py - Pa.y;
        float p2 = -(Pa.z * dx * dx + Pb.x * dy * dy + Pa.w * dx * dy);
        p2 = fminf(p2, 0.0f);
        return Pb.y * __builtin_amdgcn_exp2f(p2);
    };

#pragma unroll
    for (int k0 = 0; k0 < G; k0 += 4) {
        // ISA layout for 16x16x4 f32: lane half h supplies K = k0+2h, k0+2h+1
        const int g0 = k0 + 2 * hf;
        const float4 Pa0 = P4[g0 * 2 + 0];
        const float4 Pb0 = P4[g0 * 2 + 1];
        const float4 Pa1 = P4[g0 * 2 + 2];
        const float4 Pb1 = P4[g0 * 2 + 3];

        const v2f Bv = { sC[g0 * 4 + nc], sC[(g0 + 1) * 4 + nc] };

        const v2f A0 = { alphaf(Pa0, Pb0, py0), alphaf(Pa1, Pb1, py0) };
        const v2f A1 = { alphaf(Pa0, Pb0, py1), alphaf(Pa1, Pb1, py1) };

        // (neg_a, A, neg_b, B, c_mod, C, reuse_a, reuse_b)
        acc0 = __builtin_amdgcn_wmma_f32_16x16x4_f32(
                   false, A0, false, Bv, (short)0, acc0, false, false);
        acc1 = __builtin_amdgcn_wmma_f32_16x16x4_f32(
                   false, A1, false, Bv, (short)0, acc1, false, false);
    }

    // ---------------- Store: channel lanes (N<3) write two float4 per row ------
    if (n < 3) {
        const int b   = patch >> 10;            // / 1024
        const int p   = patch & 1023;
        const int pn0 = p >> 5;
        const int pn1 = p & 31;
        const int xb  = 8 * hf;                 // D: VGPR j holds x = j + 8*half

        const long chanRow = ((long)(b * 3 + n) * IMG) + (long)pn0 * PS;
        float* o0 = out + ((chanRow + y0) * IMG + (long)pn1 * PS + xb);
        float* o1 = out + ((chanRow + y1) * IMG + (long)pn1 * PS + xb);

        ((float4*)o0)[0] = make_float4(acc0[0], acc0[1], acc0[2], acc0[3]);
        ((float4*)o0)[1] = make_float4(acc0[4], acc0[5], acc0[6], acc0[7]);
        ((float4*)o1)[0] = make_float4(acc1[0], acc1[1], acc1[2], acc1[3]);
        ((float4*)o1)[1] = make_float4(acc1[4], acc1[5], acc1[6], acc1[7]);
    }
}

extern "C" void kernel_launch(void* const* d_in, const int* in_sizes, int n_in,
                              void* d_out, int out_size, void* d_ws, size_t ws_size,
                              hipStream_t stream) {
    const float* xyz   = (const float*)d_in[0];
    const float* chol  = (const float*)d_in[1];
    const float* color = (const float*)d_in[2];
    const float* opac  = (const float*)d_in[3];
    float* out = (float*)d_out;

    splat_wmma_kernel<<<dim3(BATCH * PATCHES), dim3(256), 0, stream>>>(
        xyz, chol, color, opac, out);
}